// GINEVirtualNodeChemBERTaClassifier_46669114638593
// MI455X (gfx1250) — compile-verified
//
#include <hip/hip_runtime.h>

#define N_NODES  100000
#define N_EDGES  1600000
#define N_GRAPHS 2048
#define IN_DIM   32
#define PEP_DIM  768
#define H        64
#define BN_EPS   1e-5f
#define MT       4            // row-tiles (of 16) per wave in big GEMM kernels
#define WPB2     2            // waves per block for MT-batched kernels
#define WPB      4            // waves per block for small WMMA kernels
#define LDS_STRIDE 65         // 16x64 tile padded to kill bank conflicts

typedef float v2f __attribute__((ext_vector_type(2)));
typedef float v8f __attribute__((ext_vector_type(8)));

// D = A(16x4) * B(4x16) + C, fp32 exact
__device__ __forceinline__ v8f wmma4(v2f a, v2f b, v8f c) {
  return __builtin_amdgcn_wmma_f32_16x16x4_f32(false, a, false, b, (short)0, c,
                                               false, false);
}

// ---------------- h = x @ in_w + in_b   [N,32] @ [32,64], 64 rows/wave --------
__global__ void __launch_bounds__(64, 1)
k_node_in(const float* __restrict__ x,
          const float* __restrict__ w,
          const float* __restrict__ bias,
          float* __restrict__ h, int nrows) {
  const int wid  = threadIdx.x >> 5;
  const int lane = threadIdx.x & 31;
  const int ntiles = nrows >> 4;
  const int njobs  = (ntiles + MT - 1) / MT;
  const int job = blockIdx.x * WPB2 + wid;
  if (job >= njobs) return;
  const int tile0 = job * MT;
  const int lo = lane & 15, hi = lane >> 4;

  int rowb[MT]; bool valid[MT];
  #pragma unroll
  for (int m = 0; m < MT; ++m) {
    const int t = tile0 + m;
    valid[m] = (t < ntiles);
    rowb[m] = (valid[m] ? t : (ntiles - 1)) << 4;
  }

  v8f acc[MT][4];
  #pragma unroll
  for (int m = 0; m < MT; ++m)
    #pragma unroll
    for (int t = 0; t < 4; ++t)
      #pragma unroll
      for (int r = 0; r < 8; ++r) acc[m][t][r] = 0.f;

  #pragma unroll
  for (int kk = 0; kk < IN_DIM / 4; ++kk) {
    const int k = kk * 4 + hi * 2;
    v2f a[MT];
    #pragma unroll
    for (int m = 0; m < MT; ++m) {
      const float* ap = x + (size_t)(rowb[m] + lo) * IN_DIM + k;
      a[m].x = ap[0]; a[m].y = ap[1];
    }
    #pragma unroll
    for (int t = 0; t < 4; ++t) {
      v2f b;
      b.x = w[(k    ) * H + t * 16 + lo];
      b.y = w[(k + 1) * H + t * 16 + lo];
      #pragma unroll
      for (int m = 0; m < MT; ++m) acc[m][t] = wmma4(a[m], b, acc[m][t]);
    }
  }
  #pragma unroll
  for (int t = 0; t < 4; ++t) {
    const int n = t * 16 + lo;
    const float bv = bias[n];
    #pragma unroll
    for (int m = 0; m < MT; ++m) {
      if (valid[m]) {
        #pragma unroll
        for (int r = 0; r < 8; ++r)
          h[(size_t)(rowb[m] + r + 8 * hi) * H + n] = acc[m][t][r] + bv;
      }
    }
  }
}

// ---------------- h += vn[batch_idx] ----------------
__global__ void k_add_vn(float* __restrict__ h, const float* __restrict__ vn,
                         const int* __restrict__ batch, int total) {
  int i = blockIdx.x * blockDim.x + threadIdx.x;
  if (i >= total) return;
  h[i] += vn[batch[i >> 6] * H + (i & 63)];
}

// ---- fused edge message: agg[dst] += relu(h[src] + ea@e_w + e_b)  (agg pre-seeded with h) ----
__global__ void k_message(const float* __restrict__ h,
                          const int* __restrict__ ei,
                          const float* __restrict__ ea,
                          const float* __restrict__ ew,
                          const float* __restrict__ eb,
                          float* __restrict__ agg) {
  int gid = blockIdx.x * blockDim.x + threadIdx.x;
  int e = gid >> 5;
  if (e >= N_EDGES) return;
  int lane = gid & 31;
  int c0 = lane * 2;
  int s = ei[e], d = ei[N_EDGES + e];
  float a0 = ea[2 * e], a1 = ea[2 * e + 1];
  float2 hv = *(const float2*)(h + (size_t)s * H + c0);
  float m0 = fmaxf(hv.x + a0 * ew[c0]     + a1 * ew[H + c0]     + eb[c0],     0.f);
  float m1 = fmaxf(hv.y + a0 * ew[c0 + 1] + a1 * ew[H + c0 + 1] + eb[c0 + 1], 0.f);
  atomicAdd(&agg[(size_t)d * H + c0],     m0);
  atomicAdd(&agg[(size_t)d * H + c0 + 1], m1);
}

// -------- node MLP: h = relu(BN( relu(z@W1+b1) @ W2 + b2 )), 64 rows/wave --------
__global__ void __launch_bounds__(64, 1)
k_node_mlp(const float* __restrict__ z,
           const float* __restrict__ W1, const float* __restrict__ B1,
           const float* __restrict__ W2, const float* __restrict__ B2,
           const float* __restrict__ bnw, const float* __restrict__ bnb,
           const float* __restrict__ bnm, const float* __restrict__ bnv,
           float* __restrict__ hout, int nrows) {
  __shared__ float tile[WPB2][MT][16 * LDS_STRIDE];   // 33 KB/block
  const int wid  = threadIdx.x >> 5;
  const int lane = threadIdx.x & 31;
  const int ntiles = nrows >> 4;
  const int njobs  = (ntiles + MT - 1) / MT;
  const int job = blockIdx.x * WPB2 + wid;
  if (job >= njobs) return;
  const int tile0 = job * MT;
  const int lo = lane & 15, hi = lane >> 4;

  int rowb[MT]; bool valid[MT];
  #pragma unroll
  for (int m = 0; m < MT; ++m) {
    const int t = tile0 + m;
    valid[m] = (t < ntiles);
    rowb[m] = (valid[m] ? t : (ntiles - 1)) << 4;
  }

  // GEMM1: acc = z @ W1 (B frag reused across 4 row tiles)
  v8f acc[MT][4];
  #pragma unroll
  for (int m = 0; m < MT; ++m)
    #pragma unroll
    for (int t = 0; t < 4; ++t)
      #pragma unroll
      for (int r = 0; r < 8; ++r) acc[m][t][r] = 0.f;

  #pragma unroll
  for (int kk = 0; kk < H / 4; ++kk) {
    const int k = kk * 4 + hi * 2;
    v2f a[MT];
    #pragma unroll
    for (int m = 0; m < MT; ++m) {
      const float* ap = z + (size_t)(rowb[m] + lo) * H + k;
      a[m].x = ap[0]; a[m].y = ap[1];
    }
    #pragma unroll
    for (int t = 0; t < 4; ++t) {
      v2f b;
      b.x = W1[(k    ) * H + t * 16 + lo];
      b.y = W1[(k + 1) * H + t * 16 + lo];
      #pragma unroll
      for (int m = 0; m < MT; ++m) acc[m][t] = wmma4(a[m], b, acc[m][t]);
    }
  }
  // bias + relu -> LDS tiles
  #pragma unroll
  for (int t = 0; t < 4; ++t) {
    const int n = t * 16 + lo;
    const float bv = B1[n];
    #pragma unroll
    for (int m = 0; m < MT; ++m)
      #pragma unroll
      for (int r = 0; r < 8; ++r)
        tile[wid][m][(r + 8 * hi) * LDS_STRIDE + n] = fmaxf(acc[m][t][r] + bv, 0.f);
  }
  asm volatile("s_wait_dscnt 0x0" ::: "memory");  // per-wave LDS RAW ordering

  // GEMM2 from LDS (acc registers recycled)
  v8f acc2[MT][4];
  #pragma unroll
  for (int m = 0; m < MT; ++m)
    #pragma unroll
    for (int t = 0; t < 4; ++t)
      #pragma unroll
      for (int r = 0; r < 8; ++r) acc2[m][t][r] = 0.f;

  #pragma unroll
  for (int kk = 0; kk < H / 4; ++kk) {
    const int k = kk * 4 + hi * 2;
    v2f a[MT];
    #pragma unroll
    for (int m = 0; m < MT; ++m) {
      a[m].x = tile[wid][m][lo * LDS_STRIDE + k];
      a[m].y = tile[wid][m][lo * LDS_STRIDE + k + 1];
    }
    #pragma unroll
    for (int t = 0; t < 4; ++t) {
      v2f b;
      b.x = W2[(k    ) * H + t * 16 + lo];
      b.y = W2[(k + 1) * H + t * 16 + lo];
      #pragma unroll
      for (int m = 0; m < MT; ++m) acc2[m][t] = wmma4(a[m], b, acc2[m][t]);
    }
  }
  // bias2 + BatchNorm(eval) + relu -> hout
  #pragma unroll
  for (int t = 0; t < 4; ++t) {
    const int n = t * 16 + lo;
    const float scale = bnw[n] * rsqrtf(bnv[n] + BN_EPS);
    const float shift = bnb[n] - bnm[n] * scale;
    const float bv = B2[n];
    #pragma unroll
    for (int m = 0; m < MT; ++m) {
      if (valid[m]) {
        #pragma unroll
        for (int r = 0; r < 8; ++r) {
          float v = (acc2[m][t][r] + bv) * scale + shift;
          hout[(size_t)(rowb[m] + r + 8 * hi) * H + n] = fmaxf(v, 0.f);
        }
      }
    }
  }
}

// -------- scatter: dst[batch[node]] += h[node] --------
__global__ void k_pool(const float* __restrict__ h, const int* __restrict__ batch,
                       float* __restrict__ dst, int total) {
  int i = blockIdx.x * blockDim.x + threadIdx.x;
  if (i >= total) return;
  atomicAdd(&dst[batch[i >> 6] * H + (i & 63)], h[i]);
}

__global__ void k_counts(const int* __restrict__ batch, float* __restrict__ counts,
                         int n) {
  int i = blockIdx.x * blockDim.x + threadIdx.x;
  if (i < n) atomicAdd(&counts[batch[i]], 1.0f);
}

// -------- vn += relu(pooled@W1+b1)@W2 + b2  (tiny: 2048 rows) --------
__global__ void k_vn_mlp(const float* __restrict__ pooled,
                         const float* __restrict__ W1, const float* __restrict__ B1,
                         const float* __restrict__ W2, const float* __restrict__ B2,
                         float* vn, int nrows) {
  __shared__ float tile[WPB][16 * LDS_STRIDE];
  const int wid  = threadIdx.x >> 5;
  const int lane = threadIdx.x & 31;
  const int tidx = blockIdx.x * WPB + wid;
  if (tidx >= (nrows >> 4)) return;
  const int row0 = tidx << 4;
  const int lo = lane & 15, hi = lane >> 4;
  float* T = &tile[wid][0];

  v8f acc[4];
  #pragma unroll
  for (int t = 0; t < 4; ++t)
    #pragma unroll
    for (int r = 0; r < 8; ++r) acc[t][r] = 0.f;

  #pragma unroll
  for (int kk = 0; kk < H / 4; ++kk) {
    const int k = kk * 4 + hi * 2;
    const size_t ai = (size_t)(row0 + lo) * H + k;
    v2f a; a.x = pooled[ai]; a.y = pooled[ai + 1];
    #pragma unroll
    for (int t = 0; t < 4; ++t) {
      v2f b;
      b.x = W1[(k    ) * H + t * 16 + lo];
      b.y = W1[(k + 1) * H + t * 16 + lo];
      acc[t] = wmma4(a, b, acc[t]);
    }
  }
  #pragma unroll
  for (int t = 0; t < 4; ++t) {
    const int n = t * 16 + lo;
    const float bv = B1[n];
    #pragma unroll
    for (int r = 0; r < 8; ++r)
      T[(r + 8 * hi) * LDS_STRIDE + n] = fmaxf(acc[t][r] + bv, 0.f);
  }
  asm volatile("s_wait_dscnt 0x0" ::: "memory");

  v8f acc2[4];
  #pragma unroll
  for (int t = 0; t < 4; ++t)
    #pragma unroll
    for (int r = 0; r < 8; ++r) acc2[t][r] = 0.f;

  #pragma unroll
  for (int kk = 0; kk < H / 4; ++kk) {
    const int k = kk * 4 + hi * 2;
    v2f a;
    a.x = T[lo * LDS_STRIDE + k];
    a.y = T[lo * LDS_STRIDE + k + 1];
    #pragma unroll
    for (int t = 0; t < 4; ++t) {
      v2f b;
      b.x = W2[(k    ) * H + t * 16 + lo];
      b.y = W2[(k + 1) * H + t * 16 + lo];
      acc2[t] = wmma4(a, b, acc2[t]);
    }
  }
  #pragma unroll
  for (int t = 0; t < 4; ++t) {
    const int n = t * 16 + lo;
    const float bv = B2[n];
    #pragma unroll
    for (int r = 0; r < 8; ++r)
      vn[(size_t)(row0 + r + 8 * hi) * H + n] += acc2[t][r] + bv;
  }
}

// -------- pep_hidden = relu(peptide_emb @ pep_w + pep_b)  K=768, 64 rows/wave --------
__global__ void __launch_bounds__(64, 1)
k_pep(const float* __restrict__ pe,
      const float* __restrict__ w, const float* __restrict__ bias,
      float* __restrict__ outp, int nrows) {
  const int wid  = threadIdx.x >> 5;
  const int lane = threadIdx.x & 31;
  const int ntiles = nrows >> 4;
  const int njobs  = (ntiles + MT - 1) / MT;
  const int job = blockIdx.x * WPB2 + wid;
  if (job >= njobs) return;
  const int tile0 = job * MT;
  const int lo = lane & 15, hi = lane >> 4;

  int rowb[MT]; bool valid[MT];
  #pragma unroll
  for (int m = 0; m < MT; ++m) {
    const int t = tile0 + m;
    valid[m] = (t < ntiles);
    rowb[m] = (valid[m] ? t : (ntiles - 1)) << 4;
  }

  v8f acc[MT][4];
  #pragma unroll
  for (int m = 0; m < MT; ++m)
    #pragma unroll
    for (int t = 0; t < 4; ++t)
      #pragma unroll
      for (int r = 0; r < 8; ++r) acc[m][t][r] = 0.f;

  #pragma unroll 2
  for (int kk = 0; kk < PEP_DIM / 4; ++kk) {
    const int k = kk * 4 + hi * 2;
    v2f a[MT];
    #pragma unroll
    for (int m = 0; m < MT; ++m) {
      const float* ap = pe + (size_t)(rowb[m] + lo) * PEP_DIM + k;
      a[m].x = ap[0]; a[m].y = ap[1];
    }
    #pragma unroll
    for (int t = 0; t < 4; ++t) {
      v2f b;
      b.x = w[(k    ) * H + t * 16 + lo];
      b.y = w[(k + 1) * H + t * 16 + lo];
      #pragma unroll
      for (int m = 0; m < MT; ++m) acc[m][t] = wmma4(a[m], b, acc[m][t]);
    }
  }
  #pragma unroll
  for (int t = 0; t < 4; ++t) {
    const int n = t * 16 + lo;
    const float bv = bias[n];
    #pragma unroll
    for (int m = 0; m < MT; ++m) {
      if (valid[m]) {
        #pragma unroll
        for (int r = 0; r < 8; ++r)
          outp[(size_t)(rowb[m] + r + 8 * hi) * H + n] = fmaxf(acc[m][t][r] + bv, 0.f);
      }
    }
  }
}

// -------- classifier: logits = relu([gsum/cnt, pep]@W1+b1) @ w2 + b2 --------
__global__ void k_cls(const float* __restrict__ gsum, const float* __restrict__ counts,
                      const float* __restrict__ pephid,
                      const float* __restrict__ W1, const float* __restrict__ B1,
                      const float* __restrict__ w2, const float* __restrict__ b2,
                      float* __restrict__ out, int ngraphs) {
  __shared__ float tile[WPB][16 * LDS_STRIDE];
  const int wid  = threadIdx.x >> 5;
  const int lane = threadIdx.x & 31;
  const int tidx = blockIdx.x * WPB + wid;
  if (tidx >= (ngraphs >> 4)) return;
  const int g0 = tidx << 4;
  const int lo = lane & 15, hi = lane >> 4;
  float* T = &tile[wid][0];

  const float invc = 1.f / fmaxf(counts[g0 + lo], 1.f);

  v8f acc[4];
  #pragma unroll
  for (int t = 0; t < 4; ++t)
    #pragma unroll
    for (int r = 0; r < 8; ++r) acc[t][r] = 0.f;

  #pragma unroll
  for (int kk = 0; kk < (2 * H) / 4; ++kk) {     // K = 128
    const int k = kk * 4 + hi * 2;
    v2f a;
    if (kk < 16) {                               // graph mean part
      const size_t ai = (size_t)(g0 + lo) * H + k;
      a.x = gsum[ai] * invc; a.y = gsum[ai + 1] * invc;
    } else {                                     // peptide hidden part
      const size_t ai = (size_t)(g0 + lo) * H + (k - H);
      a.x = pephid[ai]; a.y = pephid[ai + 1];
    }
    #pragma unroll
    for (int t = 0; t < 4; ++t) {
      v2f b;
      b.x = W1[(k    ) * H + t * 16 + lo];
      b.y = W1[(k + 1) * H + t * 16 + lo];
      acc[t] = wmma4(a, b, acc[t]);
    }
  }
  #pragma unroll
  for (int t = 0; t < 4; ++t) {
    const int n = t * 16 + lo;
    const float bv = B1[n];
    #pragma unroll
    for (int r = 0; r < 8; ++r)
      T[(r + 8 * hi) * LDS_STRIDE + n] = fmaxf(acc[t][r] + bv, 0.f);
  }
  asm volatile("s_wait_dscnt 0x0" ::: "memory");

  if (lane < 16) {
    float s = b2[0];
    #pragma unroll 8
    for (int n = 0; n < H; ++n) s += T[lo * LDS_STRIDE + n] * w2[n];
    out[g0 + lo] = s;
  }
}

extern "C" void kernel_launch(void* const* d_in, const int* in_sizes, int n_in,
                              void* d_out, int out_size, void* d_ws, size_t ws_size,
                              hipStream_t stream) {
  const float* x       = (const float*)d_in[0];
  const int*   eidx    = (const int*)  d_in[1];
  const int*   batch   = (const int*)  d_in[2];
  const float* eattr   = (const float*)d_in[3];
  const float* pep     = (const float*)d_in[4];
  const float* in_w    = (const float*)d_in[5];
  const float* in_b    = (const float*)d_in[6];
  const float* e_w     = (const float*)d_in[7];
  const float* e_b     = (const float*)d_in[8];
  const float* conv_w1 = (const float*)d_in[9];
  const float* conv_b1 = (const float*)d_in[10];
  const float* conv_w2 = (const float*)d_in[11];
  const float* conv_b2 = (const float*)d_in[12];
  const float* bn_w    = (const float*)d_in[13];
  const float* bn_b    = (const float*)d_in[14];
  const float* bn_mean = (const float*)d_in[15];
  const float* bn_var  = (const float*)d_in[16];
  const float* vn_w1   = (const float*)d_in[17];
  const float* vn_b1   = (const float*)d_in[18];
  const float* vn_w2   = (const float*)d_in[19];
  const float* vn_b2   = (const float*)d_in[20];
  const float* pep_w   = (const float*)d_in[21];
  const float* pep_b   = (const float*)d_in[22];
  const float* cls_w1  = (const float*)d_in[23];
  const float* cls_b1  = (const float*)d_in[24];
  const float* cls_w2  = (const float*)d_in[25];
  const float* cls_b2  = (const float*)d_in[26];
  float* out = (float*)d_out;

  // workspace layout (floats): ~54 MB total
  float* h      = (float*)d_ws;                       // N*H
  float* agg    = h      + (size_t)N_NODES  * H;      // N*H   (becomes z = h + msgs)
  float* vn     = agg    + (size_t)N_NODES  * H;      // G*H
  float* pooled = vn     + (size_t)N_GRAPHS * H;      // G*H
  float* pephid = pooled + (size_t)N_GRAPHS * H;      // G*H
  float* gsum   = pephid + (size_t)N_GRAPHS * H;      // G*H
  float* counts = gsum   + (size_t)N_GRAPHS * H;      // G

  const int elem_total = N_NODES * H;
  const int ntiles  = N_NODES / 16;                   // 6250
  const int njobs   = (ntiles + MT - 1) / MT;         // 1563
  const dim3 blk64(64), blk128(128), blk256(256);
  const dim3 grd_big((njobs + WPB2 - 1) / WPB2);      // 782
  const dim3 grd_elem((elem_total + 255) / 256);      // 25000
  const dim3 grd_msg((N_EDGES * 32) / 256);           // 200000
  const dim3 grd_g(N_GRAPHS / 16 / WPB);              // 32
  const dim3 grd_pep(((N_GRAPHS / 16 + MT - 1) / MT + WPB2 - 1) / WPB2);  // 16

  hipMemsetAsync(vn, 0, (size_t)N_GRAPHS * H * sizeof(float), stream);

  k_node_in<<<grd_big, blk64, 0, stream>>>(x, in_w, in_b, h, N_NODES);

  for (int i = 0; i < 3; ++i) {
    k_add_vn<<<grd_elem, blk256, 0, stream>>>(h, vn, batch, elem_total);
    // seed agg with h so message atomics produce z = h + sum(relu(h_src + e))
    hipMemcpyAsync(agg, h, (size_t)elem_total * sizeof(float),
                   hipMemcpyDeviceToDevice, stream);
    k_message<<<grd_msg, blk256, 0, stream>>>(h, eidx, eattr, e_w, e_b, agg);
    k_node_mlp<<<grd_big, blk64, 0, stream>>>(agg,
        conv_w1 + (size_t)i * H * H, conv_b1 + i * H,
        conv_w2 + (size_t)i * H * H, conv_b2 + i * H,
        bn_w + i * H, bn_b + i * H, bn_mean + i * H, bn_var + i * H,
        h, N_NODES);
    if (i != 2) {
      hipMemsetAsync(pooled, 0, (size_t)N_GRAPHS * H * sizeof(float), stream);
      k_pool<<<grd_elem, blk256, 0, stream>>>(h, batch, pooled, elem_total);
      k_vn_mlp<<<grd_g, blk128, 0, stream>>>(pooled, vn_w1, vn_b1, vn_w2, vn_b2,
                                             vn, N_GRAPHS);
    }
  }

  // gsum + counts are contiguous -> one memset
  hipMemsetAsync(gsum, 0, (size_t)(N_GRAPHS * H + N_GRAPHS) * sizeof(float), stream);
  k_pool<<<grd_elem, blk256, 0, stream>>>(h, batch, gsum, elem_total);
  k_counts<<<dim3((N_NODES + 255) / 256), blk256, 0, stream>>>(batch, counts, N_NODES);

  k_pep<<<grd_pep, blk64, 0, stream>>>(pep, pep_w, pep_b, pephid, N_GRAPHS);
  k_cls<<<grd_g, blk128, 0, stream>>>(gsum, counts, pephid,
                                      cls_w1, cls_b1, cls_w2, cls_b2, out, N_GRAPHS);
}